// SpatialTransformer_34342558499003
// MI455X (gfx1250) — compile-verified
//
#include <hip/hip_runtime.h>

// Spatial transformer (affine grid + bilinear sample), MI455X / gfx1250.
// Memory-bound: ~600 MB traffic @ 23.3 TB/s => ~26 us floor. fp32 throughout.
// - Affine-grid einsum computed with V_WMMA_F32_16X16X4_F32 (2 wmma/wave,
//   delivering each lane its own (x,y) sample coordinate in-register).
// - Gather: 4 corners x 32ch via fully-unrolled global_load_b128.
// - Output stores are non-temporal (TH=NT): output is write-once/never-read,
//   keep the 192MB L2 for gather-corner reuse of the input image instead.

typedef __attribute__((ext_vector_type(2))) float v2f;
typedef __attribute__((ext_vector_type(4))) float f4;
typedef __attribute__((ext_vector_type(8))) float v8f;

#define BN 16
#define HN 384
#define WN 384
#define CN 32
// pixels per batch = 384*384 = 147456 ; blocks per batch = 147456/256 = 576

__global__ __launch_bounds__(256) void SpatialTransformer_34342558499003_kernel(
    const float* __restrict__ img,    // [B,H,W,C]
    const float* __restrict__ theta,  // [B,6]
    float* __restrict__ out)          // [B,H,W,C]
{
    const int tid      = threadIdx.x;
    const int lane     = tid & 31;
    const int pix      = blockIdx.x * 256 + tid;   // global pixel id (exact cover)
    const int wavebase = pix & ~31;
    const int b        = blockIdx.x / 576;         // uniform per block -> SALU

    // ---- per-batch adjusted affine params (scalar path) ----
    const float* t = theta + b * 6;
    const float t00 = t[0], t01 = t[1], t02 = t[2];
    const float t10 = t[3], t11 = t[4], t12 = t[5];
    const float rdet = 1.0f / (t00 * t11 - t01 * t10);
    const float m00 =  t11 * rdet, m01 = -t01 * rdet;
    const float m10 = -t10 * rdet, m11 =  t00 * rdet;
    const float xc = (0.5f - t02) * m00;   // x_center
    const float yc = (0.5f - t12) * m11;   // y_center
    // Fold (g+1)*dim/2 into the affine rows:
    //   x = 0.5*W*m00*xg + 0.5*W*m01*yg + W*yc
    //   y = 0.5*H*m10*xg + 0.5*H*m11*yg + H*xc
    const float Ax0 = 0.5f * WN * m00, Ax1 = 0.5f * WN * m01, Cx = (float)WN * yc;
    const float Ay0 = 0.5f * HN * m10, Ay1 = 0.5f * HN * m11, Cy = (float)HN * xc;

    // ---- WMMA fragments (V_WMMA_F32_16X16X4_F32) ----
    // A (16x4, 2 VGPRs): lanes0-15 hold A[M=lane][K=0..1], lanes16-31 A[M][K=2..3]
    // rows 0,8 -> x transform ; rows 1,9 -> y transform ; others zero.
    const int  l15 = lane & 15;
    const bool lo  = lane < 16;
    const bool isx = (l15 == 0) || (l15 == 8);
    const bool isy = (l15 == 1) || (l15 == 9);

    v2f a;
    a.x = lo ? (isx ? Ax0 : (isy ? Ay0 : 0.0f))
             : (isx ? Cx  : (isy ? Cy  : 0.0f));
    a.y = lo ? (isx ? Ax1 : (isy ? Ay1 : 0.0f)) : 0.0f;

    // B (4x16, 2 VGPRs): lanes0-15 hold B[K=0..1][N=lane] = (xg,yg) of pixel,
    // lanes16-31 hold B[K=2..3][N] = (1,0).
    const int p0 = wavebase + l15;        // columns for wmma#0: pixels +0..15
    const int p1 = p0 + 16;               // columns for wmma#1: pixels +16..31
    const int w0 = p0 % WN, h0 = (p0 / WN) % HN;
    const int w1 = p1 % WN, h1 = (p1 / WN) % HN;
    const float sx = 2.0f / (float)(WN - 1);
    const float sy = 2.0f / (float)(HN - 1);
    v2f b0, b1;
    b0.x = lo ? ((float)w0 * sx - 1.0f) : 1.0f;
    b0.y = lo ? ((float)h0 * sy - 1.0f) : 0.0f;
    b1.x = lo ? ((float)w1 * sx - 1.0f) : 1.0f;
    b1.y = lo ? ((float)h1 * sy - 1.0f) : 0.0f;

    v8f z = {};
    v8f d0 = __builtin_amdgcn_wmma_f32_16x16x4_f32(false, a, false, b0,
                                                   (short)0, z, false, false);
    v8f d1 = __builtin_amdgcn_wmma_f32_16x16x4_f32(false, a, false, b1,
                                                   (short)0, z, false, false);

    // lanes 0-15: D[0][n]=x, D[1][n]=y (vgpr0/1). lanes 16-31: D[8][n], D[9][n].
    const float x = lo ? d0[0] : d1[0];
    const float y = lo ? d0[1] : d1[1];

    // ---- bilinear weights (match reference: clamp, then subtract) ----
    const int x0 = (int)floorf(x), y0 = (int)floorf(y);
    const int x0c = min(max(x0,     0), WN - 1);
    const int x1c = min(max(x0 + 1, 0), WN - 1);
    const int y0c = min(max(y0,     0), HN - 1);
    const int y1c = min(max(y0 + 1, 0), HN - 1);
    const float x0f = (float)x0c, x1f = (float)x1c;
    const float y0f = (float)y0c, y1f = (float)y1c;
    const float wa = (x1f - x) * (y1f - y);
    const float wb = (x1f - x) * (y - y0f);
    const float wc = (x - x0f) * (y1f - y);
    const float wd = (x - x0f) * (y - y0f);

    // ---- gather + blend: 4 corners x 32ch, b128 vector loads/stores ----
    const size_t ib = (size_t)b * ((size_t)HN * WN * CN);
    const float* pA = img + ib + ((size_t)((y0c * WN) + x0c)) * CN;
    const float* pB = img + ib + ((size_t)((y1c * WN) + x0c)) * CN;
    const float* pC = img + ib + ((size_t)((y0c * WN) + x1c)) * CN;
    const float* pD = img + ib + ((size_t)((y1c * WN) + x1c)) * CN;
    float*       pO = out + (size_t)pix * CN;

#pragma unroll
    for (int c = 0; c < CN; c += 4) {
        f4 ga = *(const f4*)(pA + c);
        f4 gb = *(const f4*)(pB + c);
        f4 gc = *(const f4*)(pC + c);
        f4 gd = *(const f4*)(pD + c);
        f4 o  = ga * wa;
        o += gb * wb;
        o += gc * wc;
        o += gd * wd;
        __builtin_nontemporal_store(o, (f4*)(pO + c));   // TH=NT: don't pollute L2
    }
}

extern "C" void kernel_launch(void* const* d_in, const int* in_sizes, int n_in,
                              void* d_out, int out_size, void* d_ws, size_t ws_size,
                              hipStream_t stream) {
    const float* images = (const float*)d_in[0];   // (16,384,384,32) f32
    const float* theta  = (const float*)d_in[1];   // (16,6) f32
    float* out          = (float*)d_out;

    const int total_pixels = BN * HN * WN;         // 2,359,296
    const int blocks = total_pixels / 256;         // 9216 (exact)
    SpatialTransformer_34342558499003_kernel<<<blocks, 256, 0, stream>>>(images, theta, out);
}